// BD_12335146074506
// MI455X (gfx1250) — compile-verified
//
#include <hip/hip_runtime.h>

typedef __attribute__((ext_vector_type(2))) float v2f;
typedef __attribute__((ext_vector_type(8))) float v8f;

namespace {
constexpr int N_IN  = 64;
constexpr int HID   = 32;
constexpr int N_OUT = 24;
constexpr int NLAY  = 4;
constexpr int HSTR  = 34;   // padded LDS row stride (floats): even (b64 align) + bank-spread
constexpr int WAVES = 8;    // 256 threads / wave32

__device__ inline v8f wmma4(v2f a, v2f b, v8f c) {
  // D = A(16x4 f32) * B(4x16 f32) + C(16x16 f32)
  return __builtin_amdgcn_wmma_f32_16x16x4_f32(false, a, false, b, (short)0, c,
                                               false, false);
}

__device__ inline v8f relu8(v8f c) {
#pragma unroll
  for (int i = 0; i < 8; ++i) c[i] = fmaxf(c[i], 0.0f);
  return c;
}
}  // namespace

__global__ __launch_bounds__(256) void bd_mlp_wmma(
    const float* __restrict__ x,     // [B][64]
    const float* __restrict__ win,   // [64][32]
    const float* __restrict__ wh,    // [4][32][32]
    const float* __restrict__ wo,    // [32][24]
    float* __restrict__ out,         // [B][24]
    int nrows) {
  __shared__ float hbuf[WAVES][16 * HSTR];

  const int lane = threadIdx.x & 31;
  const int wave = threadIdx.x >> 5;
  const int row0 = (blockIdx.x * WAVES + wave) * 16;
  if (row0 >= nrows) return;  // wave-uniform: EXEC stays all-ones for WMMA

  const int mn  = lane & 15;         // A-row / B-col / C-col index for this lane
  const int kk  = (lane >> 4) << 1;  // K pair offset within a 4-wide K step
  const int mhi = (lane >> 4) << 3;  // +8 C/D row offset for upper lane half
  float* hb = hbuf[wave];

  // ---------------- input layer: h = relu(x @ Win), dense K=64 ----------------
  v8f c0 = {};
  v8f c1 = {};
  const float* xr = x + (size_t)(row0 + mn) * N_IN;
#pragma unroll
  for (int k0 = 0; k0 < N_IN; k0 += 4) {
    const int k = k0 + kk;
    v2f a;
    a.x = xr[k];
    a.y = xr[k + 1];
    v2f b0, b1;
    b0.x = win[k * HID + mn];
    b0.y = win[(k + 1) * HID + mn];
    b1.x = win[k * HID + 16 + mn];
    b1.y = win[(k + 1) * HID + 16 + mn];
    c0 = wmma4(a, b0, c0);
    c1 = wmma4(a, b1, c1);
  }
  c0 = relu8(c0);
  c1 = relu8(c1);
#pragma unroll
  for (int v = 0; v < 8; ++v) {
    hb[(v + mhi) * HSTR + mn]      = c0[v];
    hb[(v + mhi) * HSTR + 16 + mn] = c1[v];
  }

  // ------- hidden layers: block-diag (4x 8x8). Cols 0..15 <- rows 0..15 only,
  // ------- cols 16..31 <- rows 16..31 only => K=16 per N-tile.
#pragma unroll
  for (int L = 0; L < NLAY; ++L) {
    const float* w = wh + L * HID * HID;
    v8f d0 = {};
    v8f d1 = {};
#pragma unroll
    for (int kb = 0; kb < 16; kb += 4) {
      {  // N-tile 0
        const int k = kb + kk;  // 0..15
        v2f a;
        a.x = hb[mn * HSTR + k];
        a.y = hb[mn * HSTR + k + 1];
        v2f b;
        b.x = ((k >> 3) == (mn >> 3))       ? w[k * HID + mn]       : 0.0f;
        b.y = (((k + 1) >> 3) == (mn >> 3)) ? w[(k + 1) * HID + mn] : 0.0f;
        d0 = wmma4(a, b, d0);
      }
      {  // N-tile 1
        const int k = 16 + kb + kk;  // 16..31
        const int n = 16 + mn;
        v2f a;
        a.x = hb[mn * HSTR + k];
        a.y = hb[mn * HSTR + k + 1];
        v2f b;
        b.x = ((k >> 3) == (n >> 3))       ? w[k * HID + n]       : 0.0f;
        b.y = (((k + 1) >> 3) == (n >> 3)) ? w[(k + 1) * HID + n] : 0.0f;
        d1 = wmma4(a, b, d1);
      }
    }
    d0 = relu8(d0);
    d1 = relu8(d1);
#pragma unroll
    for (int v = 0; v < 8; ++v) {
      hb[(v + mhi) * HSTR + mn]      = d0[v];
      hb[(v + mhi) * HSTR + 16 + mn] = d1[v];
    }
  }

  // ------- output layer: block-diag (8x 4x3), no activation.
  // cols 0..15 need rows 0..23; cols 16..23 need rows 20..31.
  v8f o0 = {};
  v8f o1 = {};
#pragma unroll
  for (int k0 = 0; k0 < 24; k0 += 4) {
    const int k = k0 + kk;
    v2f a;
    a.x = hb[mn * HSTR + k];
    a.y = hb[mn * HSTR + k + 1];
    v2f b;
    b.x = ((k >> 2) == (mn / 3))       ? wo[k * N_OUT + mn]       : 0.0f;
    b.y = (((k + 1) >> 2) == (mn / 3)) ? wo[(k + 1) * N_OUT + mn] : 0.0f;
    o0 = wmma4(a, b, o0);
  }
#pragma unroll
  for (int k0 = 20; k0 < 32; k0 += 4) {
    const int k = k0 + kk;
    const int n = 16 + mn;
    v2f a;
    a.x = hb[mn * HSTR + k];
    a.y = hb[mn * HSTR + k + 1];
    v2f b;
    b.x = (n < N_OUT && (k >> 2) == (n / 3))       ? wo[k * N_OUT + n]       : 0.0f;
    b.y = (n < N_OUT && ((k + 1) >> 2) == (n / 3)) ? wo[(k + 1) * N_OUT + n] : 0.0f;
    o1 = wmma4(a, b, o1);
  }

  float* orow = out + (size_t)row0 * N_OUT;
#pragma unroll
  for (int v = 0; v < 8; ++v) {
    orow[(v + mhi) * N_OUT + mn] = o0[v];
    if (mn < 8) orow[(v + mhi) * N_OUT + 16 + mn] = o1[v];
  }
}

extern "C" void kernel_launch(void* const* d_in, const int* in_sizes, int n_in,
                              void* d_out, int out_size, void* d_ws, size_t ws_size,
                              hipStream_t stream) {
  const float* x   = (const float*)d_in[0];
  const float* win = (const float*)d_in[1];
  const float* wh  = (const float*)d_in[2];
  const float* wo  = (const float*)d_in[3];
  float* out = (float*)d_out;

  const int nrows  = in_sizes[0] / N_IN;           // 1048576
  const int tiles  = (nrows + 15) / 16;            // 65536 16-row tiles
  const int blocks = (tiles + WAVES - 1) / WAVES;  // 8 tiles (waves) per block
  bd_mlp_wmma<<<blocks, 256, 0, stream>>>(x, win, wh, wo, out, nrows);
}